// Muiti_PostGCN_17411797418122
// MI455X (gfx1250) — compile-verified
//
#include <hip/hip_runtime.h>

#define U_ 8192
#define P_ 4096
#define D_ 64
#define B_ 1024
#define H_ 50
#define R_ 5
#define EPS_ 1e-8f

typedef float v2f __attribute__((ext_vector_type(2)));
typedef float v4f __attribute__((ext_vector_type(4)));
typedef float v8f __attribute__((ext_vector_type(8)));

#define GLOBAL_AS __attribute__((address_space(1)))

// Force a global_load_b64 (not flat_load): flat ops tick DScnt and serialize
// against the LDS pipe on CDNA5; global ops are gated by LOADcnt only.
__device__ __forceinline__ v2f gload2(const float* p) {
    return *(const GLOBAL_AS v2f*)(unsigned long long)p;
}

// ---------------------------------------------------------------------------
// Kernel 1:  Out[r] (M x 64) = A[r] (M x 4096) @ X[r] (4096 x 64), fp32 WMMA.
// grid.x = M/128 (8 waves x 16 rows), grid.y = r.
// B operand staged in LDS pair-interleaved: lds[(k>>1)*128 + n*2 + (k&1)]
// so each 4x16 B fragment pair is a single aligned ds_load_b64.
// ---------------------------------------------------------------------------
__global__ __launch_bounds__(256)
void gemm_a_x(const float* __restrict__ A0, const float* __restrict__ A1,
              const float* __restrict__ A2, const float* __restrict__ A3,
              const float* __restrict__ A4,
              const float* __restrict__ Xbase, unsigned long long xStride,
              float* __restrict__ Obase, unsigned long long oStride)
{
    __shared__ float ldsB[64 * 128];   // 32 KB: 64 K-pairs x 64 cols x 2

    const float* As[R_] = {A0, A1, A2, A3, A4};
    const float* __restrict__ A = As[blockIdx.y];
    const float* __restrict__ X = Xbase + xStride * blockIdx.y;
    float* __restrict__ Out = Obase + oStride * blockIdx.y;

    const int tid    = threadIdx.x;
    const int wave   = tid >> 5;
    const int lane   = tid & 31;
    const int laneM  = lane & 15;
    const int laneHi = lane >> 4;          // 0: K+0/K+1, 1: K+2/K+3

    const int row0 = blockIdx.x * 128 + wave * 16;
    const float* __restrict__ Arow = A + (size_t)(row0 + laneM) * P_;

    v8f acc0 = {}, acc1 = {}, acc2 = {}, acc3 = {};

    for (int kc = 0; kc < P_; kc += 128) {
        __syncthreads();
        // Stage X[kc..kc+127][0..63] -> pair-interleaved LDS.
        // q: kpair = q>>5 (64), n2 = (q&31)*2; 2048 quads / 256 thr = 8 iters.
        #pragma unroll
        for (int it = 0; it < 8; ++it) {
            const int q     = it * 256 + tid;
            const int kpair = q >> 5;
            const int n2    = (q & 31) * 2;
            const float* s0 = X + (size_t)(kc + 2 * kpair) * D_ + n2;
            const v2f r0 = gload2(s0);
            const v2f r1 = gload2(s0 + D_);
            v4f w; w.x = r0.x; w.y = r1.x; w.z = r0.y; w.w = r1.y;
            *(v4f*)&ldsB[kpair * 128 + n2 * 2] = w;
        }
        __syncthreads();

        #pragma unroll 4
        for (int kk = 0; kk < 128; kk += 4) {
            const v2f a = gload2(Arow + kc + kk + 2 * laneHi);
            const int kp = (kk >> 1) + laneHi;          // even krow / 2
            const float* bb = &ldsB[kp * 128 + laneM * 2];
            const v2f b0 = *(const v2f*)(bb +  0);
            const v2f b1 = *(const v2f*)(bb + 32);
            const v2f b2 = *(const v2f*)(bb + 64);
            const v2f b3 = *(const v2f*)(bb + 96);

            acc0 = __builtin_amdgcn_wmma_f32_16x16x4_f32(false, a, false, b0, (short)0, acc0, false, false);
            acc1 = __builtin_amdgcn_wmma_f32_16x16x4_f32(false, a, false, b1, (short)0, acc1, false, false);
            acc2 = __builtin_amdgcn_wmma_f32_16x16x4_f32(false, a, false, b2, (short)0, acc2, false, false);
            acc3 = __builtin_amdgcn_wmma_f32_16x16x4_f32(false, a, false, b3, (short)0, acc3, false, false);
        }
    }

    #pragma unroll
    for (int v = 0; v < 8; ++v) {
        const size_t r = (size_t)(row0 + v + 8 * laneHi) * D_;
        Out[r +  0 + laneM] = acc0[v];
        Out[r + 16 + laneM] = acc1[v];
        Out[r + 32 + laneM] = acc2[v];
        Out[r + 48 + laneM] = acc3[v];
    }
}

// ---------------------------------------------------------------------------
// Kernel 2:  Out[r] (4096 x 64) = A[r]^T @ X[r] (8192 x 64).
// A tile transposed through LDS, pair-interleaved on K for ds_load_b64
// operand fetch:  ldsA[(k>>1)*256 + p*2 + (k&1)].
// grid.x = P/128, grid.y = r.
// ---------------------------------------------------------------------------
__global__ __launch_bounds__(256)
void gemm_at_x(const float* __restrict__ A0, const float* __restrict__ A1,
               const float* __restrict__ A2, const float* __restrict__ A3,
               const float* __restrict__ A4,
               const float* __restrict__ Xbase, unsigned long long xStride,
               float* __restrict__ Obase, unsigned long long oStride)
{
    __shared__ float ldsA[32 * 256];   // 32 KB: 32 K-pairs x 128 p x 2
    __shared__ float ldsX[32 * 128];   // 16 KB: 32 K-pairs x 64 n x 2

    const float* As[R_] = {A0, A1, A2, A3, A4};
    const float* __restrict__ A = As[blockIdx.y];
    const float* __restrict__ X = Xbase + xStride * blockIdx.y;
    float* __restrict__ Out = Obase + oStride * blockIdx.y;

    const int tid    = threadIdx.x;
    const int wave   = tid >> 5;
    const int lane   = tid & 31;
    const int laneM  = lane & 15;
    const int laneHi = lane >> 4;

    const int p0 = blockIdx.x * 128;
    const int pl = wave * 16 + laneM;      // local output row (p) 0..127

    v8f acc0 = {}, acc1 = {}, acc2 = {}, acc3 = {};

    for (int uc = 0; uc < U_; uc += 64) {
        __syncthreads();
        // Stage A[uc..uc+63][p0..p0+127] -> pair-interleaved ldsA.
        // q: kpair = q>>6 (32), p2 = (q&63)*2; 2048 quads -> 8 iters.
        #pragma unroll
        for (int it = 0; it < 8; ++it) {
            const int q     = it * 256 + tid;
            const int kpair = q >> 6;
            const int p2    = (q & 63) * 2;
            const float* s0 = A + (size_t)(uc + 2 * kpair) * P_ + p0 + p2;
            const v2f r0 = gload2(s0);
            const v2f r1 = gload2(s0 + P_);
            v4f w; w.x = r0.x; w.y = r1.x; w.z = r0.y; w.w = r1.y;
            *(v4f*)&ldsA[kpair * 256 + p2 * 2] = w;
        }
        // Stage X[uc..uc+63][0..63] -> pair-interleaved ldsX.
        // q: kpair = q>>5 (32), n2 = (q&31)*2; 1024 quads -> 4 iters.
        #pragma unroll
        for (int it = 0; it < 4; ++it) {
            const int q     = it * 256 + tid;
            const int kpair = q >> 5;
            const int n2    = (q & 31) * 2;
            const float* s0 = X + (size_t)(uc + 2 * kpair) * D_ + n2;
            const v2f r0 = gload2(s0);
            const v2f r1 = gload2(s0 + D_);
            v4f w; w.x = r0.x; w.y = r1.x; w.z = r0.y; w.w = r1.y;
            *(v4f*)&ldsX[kpair * 128 + n2 * 2] = w;
        }
        __syncthreads();

        #pragma unroll 4
        for (int kk = 0; kk < 64; kk += 4) {
            const int kp = (kk >> 1) + laneHi;
            const v2f a  = *(const v2f*)&ldsA[kp * 256 + pl * 2]; // A^T[p,k],A^T[p,k+1]
            const float* bb = &ldsX[kp * 128 + laneM * 2];
            const v2f b0 = *(const v2f*)(bb +  0);
            const v2f b1 = *(const v2f*)(bb + 32);
            const v2f b2 = *(const v2f*)(bb + 64);
            const v2f b3 = *(const v2f*)(bb + 96);

            acc0 = __builtin_amdgcn_wmma_f32_16x16x4_f32(false, a, false, b0, (short)0, acc0, false, false);
            acc1 = __builtin_amdgcn_wmma_f32_16x16x4_f32(false, a, false, b1, (short)0, acc1, false, false);
            acc2 = __builtin_amdgcn_wmma_f32_16x16x4_f32(false, a, false, b2, (short)0, acc2, false, false);
            acc3 = __builtin_amdgcn_wmma_f32_16x16x4_f32(false, a, false, b3, (short)0, acc3, false, false);
        }
    }

    #pragma unroll
    for (int v = 0; v < 8; ++v) {
        const size_t r = (size_t)(p0 + wave * 16 + v + 8 * laneHi) * D_;
        Out[r +  0 + laneM] = acc0[v];
        Out[r + 16 + laneM] = acc1[v];
        Out[r + 32 + laneM] = acc2[v];
        Out[r + 48 + laneM] = acc3[v];
    }
}

// ---------------------------------------------------------------------------
// Elementwise / reduction tail kernels.
// ---------------------------------------------------------------------------
__global__ __launch_bounds__(256)
void combine_u(const float* __restrict__ u0, const float* __restrict__ u1,
               const float* __restrict__ u2,
               float* __restrict__ base, float* __restrict__ rel)
{
    const int i = blockIdx.x * 256 + threadIdx.x;   // 0 .. 5*U*D-1
    const int r = i / (U_ * D_);
    const int j = i - r * (U_ * D_);
    const float v = (u0[j] + u1[i] + u2[i]) * (1.0f / 3.0f);
    if (r == 0) base[j] = v;
    else        rel[(size_t)(r - 1) * (U_ * D_) + j] = v;
}

__global__ __launch_bounds__(256)
void combine_p(const float* __restrict__ p0, const float* __restrict__ p1_0,
               const float* __restrict__ p2_0, float* __restrict__ Xpf)
{
    const int i = blockIdx.x * 256 + threadIdx.x;   // 0 .. P*D-1
    Xpf[i] = (p0[i] + p1_0[i] + p2_0[i]) * (1.0f / 3.0f);
}

__global__ __launch_bounds__(256)
void user_embed(const float* __restrict__ rel, const float* __restrict__ W,
                float* __restrict__ ue)
{
    const int i = blockIdx.x * 256 + threadIdx.x;   // 0 .. U*D-1
    const int u = i >> 6;
    float s = 0.0f;
    #pragma unroll
    for (int j = 0; j < 4; ++j)
        s += W[u * 4 + j] * rel[(size_t)j * (U_ * D_) + i];
    ue[i] = s;
}

__global__ __launch_bounds__(256)
void stats_kernel(const float* __restrict__ base, const float* __restrict__ rel,
                  float* __restrict__ var_out)
{
    const int u = blockIdx.x * 256 + threadIdx.x;   // one u per thread
    const float* __restrict__ br = base + (size_t)u * D_;

    float nb = 0.0f;
    float dot[4] = {0.f, 0.f, 0.f, 0.f};
    float nr2[4] = {0.f, 0.f, 0.f, 0.f};
    for (int d = 0; d < D_; ++d) {
        const float bv = br[d];
        nb += bv * bv;
        #pragma unroll
        for (int j = 0; j < 4; ++j) {
            const float rv = rel[(size_t)j * (U_ * D_) + (size_t)u * D_ + d];
            dot[j] += bv * rv;
            nr2[j] += rv * rv;
        }
    }
    const float n0 = sqrtf(nb);
    float sim[4];
    #pragma unroll
    for (int j = 0; j < 4; ++j)
        sim[j] = dot[j] / fmaxf(n0 * sqrtf(nr2[j]), EPS_);

    float smax = sim[0];
    #pragma unroll
    for (int j = 1; j < 4; ++j) smax = fmaxf(smax, sim[j]);

    float simm[4];
    #pragma unroll
    for (int j = 0; j < 4; ++j) simm[j] = (sim[j] == smax) ? sim[j] : 0.0f;

    const float mean = (simm[0] + simm[1] + simm[2] + simm[3]) * 0.25f;
    float s2 = 0.0f;
    #pragma unroll
    for (int j = 0; j < 4; ++j) { const float dd = simm[j] - mean; s2 += dd * dd; }
    const float sd = sqrtf(s2 * (1.0f / 3.0f));     // ddof = 1
    var_out[u] = logf(sd + 1.0f);
}

__global__ __launch_bounds__(64)
void gather_out(const float* __restrict__ Xpf, const float* __restrict__ base,
                const float* __restrict__ var, const float* __restrict__ ue,
                const int* __restrict__ src_id, const int* __restrict__ rum_id,
                float* __restrict__ out)
{
    const int b = blockIdx.x;
    const int d = threadIdx.x;

    const int sid = src_id[b];
    out[(size_t)b * D_ + d] = Xpf[(size_t)sid * D_ + d];

    float av = 0.0f, am = 0.0f, aa = 0.0f;
    for (int h = 0; h < H_; ++h) {
        const int uid = rum_id[b * H_ + h];
        const float bs = base[(size_t)uid * D_ + d];
        av += var[uid] * bs;
        am += bs;
        aa += ue[(size_t)uid * D_ + d];
    }
    out[1 * B_ * D_ + b * D_ + d] = av;                    // X_var_user
    out[2 * B_ * D_ + b * D_ + d] = aa * (1.0f / H_);      // AugUser
    out[3 * B_ * D_ + b * D_ + d] = am * (1.0f / H_);      // X_mean_user
}

// ---------------------------------------------------------------------------
extern "C" void kernel_launch(void* const* d_in, const int* in_sizes, int n_in,
                              void* d_out, int out_size, void* d_ws, size_t ws_size,
                              hipStream_t stream) {
    (void)in_sizes; (void)n_in; (void)out_size; (void)ws_size;

    const float* A0 = (const float*)d_in[0];
    const float* A1 = (const float*)d_in[1];
    const float* A2 = (const float*)d_in[2];
    const float* A3 = (const float*)d_in[3];
    const float* A4 = (const float*)d_in[4];
    const float* u0 = (const float*)d_in[5];   // user_emb  (U x D)
    const float* p0 = (const float*)d_in[6];   // post_emb  (P x D)
    const float* W  = (const float*)d_in[7];   // user_model (U x 4)
    const int* src  = (const int*)d_in[8];     // X_source_id (B)
    const int* rum  = (const int*)d_in[10];    // X_rumer_id  (B x H)

    float* ws = (float*)d_ws;
    size_t off = 0;
    float* u1   = ws + off; off += (size_t)R_ * U_ * D_;   // 5 x U x D
    float* p1   = ws + off; off += (size_t)R_ * P_ * D_;   // 5 x P x D
    float* u2   = ws + off; off += (size_t)R_ * U_ * D_;   // 5 x U x D
    float* p2   = ws + off; off += (size_t)P_ * D_;        // r=0 only
    float* base = ws + off; off += (size_t)U_ * D_;
    float* rel  = ws + off; off += (size_t)4 * U_ * D_;
    float* ue   = ws + off; off += (size_t)U_ * D_;
    float* Xpf  = ws + off; off += (size_t)P_ * D_;
    float* var  = ws + off; off += (size_t)U_;

    const dim3 blk(256);

    // Stage 1: u1[r] = A[r] @ p0
    gemm_a_x<<<dim3(U_ / 128, R_), blk, 0, stream>>>(
        A0, A1, A2, A3, A4, p0, 0ull, u1, (unsigned long long)U_ * D_);
    // Stage 2: p1[r] = A[r]^T @ u0
    gemm_at_x<<<dim3(P_ / 128, R_), blk, 0, stream>>>(
        A0, A1, A2, A3, A4, u0, 0ull, p1, (unsigned long long)P_ * D_);
    // Stage 3: u2[r] = A[r] @ p1[r]
    gemm_a_x<<<dim3(U_ / 128, R_), blk, 0, stream>>>(
        A0, A1, A2, A3, A4, p1, (unsigned long long)P_ * D_, u2,
        (unsigned long long)U_ * D_);
    // Stage 4: p2[0] = A[0]^T @ u1[0]   (only r=0 consumed downstream)
    gemm_at_x<<<dim3(P_ / 128, 1), blk, 0, stream>>>(
        A0, A0, A0, A0, A0, u1, 0ull, p2, 0ull);

    combine_u<<<(R_ * U_ * D_) / 256, blk, 0, stream>>>(u0, u1, u2, base, rel);
    combine_p<<<(P_ * D_) / 256, blk, 0, stream>>>(p0, p1, p2, Xpf);
    user_embed<<<(U_ * D_) / 256, blk, 0, stream>>>(rel, W, ue);
    stats_kernel<<<U_ / 256, blk, 0, stream>>>(base, rel, var);
    gather_out<<<B_, dim3(64), 0, stream>>>(Xpf, base, var, ue, src, rum,
                                            (float*)d_out);
}